// NetGINE_43516608643681
// MI455X (gfx1250) — compile-verified
//
#include <hip/hip_runtime.h>
#include <math.h>

// ---- problem constants (match reference) ----
#define DIMF   96
#define NFEAT  9
#define NGRAPH 1000
#define S2S_STEPS 6

typedef __attribute__((ext_vector_type(2))) float v2f;
typedef __attribute__((ext_vector_type(8))) float v8f;
typedef __attribute__((ext_vector_type(4))) unsigned int u32x4;
typedef __attribute__((ext_vector_type(8))) int i32x8;
typedef __attribute__((ext_vector_type(4))) int i32x4;

#if defined(__has_builtin)
#  if __has_builtin(__builtin_amdgcn_tensor_load_to_lds)
#    define HAVE_TDM 1
#  endif
#endif
#ifndef HAVE_TDM
#  define HAVE_TDM 0
#endif

// CDNA5 fp32 WMMA: D(16x16,f32) = A(16x4,f32) * B(4x16,f32) + C
__device__ __forceinline__ v8f wmma_f32_16x16x4(v2f a, v2f b, v8f c) {
  return __builtin_amdgcn_wmma_f32_16x16x4_f32(false, a, false, b, (short)0, c, false, false);
}

#if HAVE_TDM
// TDM: DMA a 96x96 fp32 row-major matrix from global into LDS at lds_byte_off.
// D# built per CDNA5 ISA ch.8: group0 {count=1, lds_addr, global_addr, type=2},
// group1 {data_size=4B, tensor_dim0/1=96, tile_dim0/1=96, dim0_stride=96}.
__device__ __forceinline__ void tdm_load_w96(const float* gptr, unsigned lds_byte_off) {
  unsigned long long ga = (unsigned long long)(uintptr_t)gptr;
  u32x4 g0;
  g0.x = 0x1u;                                    // count=1 (valid), user mode
  g0.y = lds_byte_off;                            // lds_addr[31:0]
  g0.z = (unsigned)(ga & 0xFFFFFFFFu);            // global_addr[31:0]
  g0.w = (unsigned)((ga >> 32) & 0x01FFFFFFu)     // global_addr[56:32]
         | (2u << 30);                            // type=2 ("image")
  i32x8 g1;
  g1[0] = (int)(2u << 16);                        // data_size=2 -> 4 bytes
  g1[1] = (int)(96u << 16);                       // tensor_dim0[15:0]=96
  g1[2] = (int)(96u << 16);                       // tensor_dim0 hi=0 | tensor_dim1 lo=96
  g1[3] = (int)(96u << 16);                       // tensor_dim1 hi=0 | tile_dim0=96
  g1[4] = 96;                                     // tile_dim1=96, tile_dim2=0
  g1[5] = 96;                                     // tensor_dim0_stride[31:0]=96
  g1[6] = 0;
  g1[7] = 0;
  i32x4 z4 = {0, 0, 0, 0};
#if defined(__clang_major__) && (__clang_major__ >= 23)
  i32x8 z8 = {0, 0, 0, 0, 0, 0, 0, 0};
  __builtin_amdgcn_tensor_load_to_lds(g0, g1, z4, z4, z8, 0);
#else
  __builtin_amdgcn_tensor_load_to_lds(g0, g1, z4, z4, 0);
#endif
}
#endif

// ------------------------------------------------------------------
// utility kernels
// ------------------------------------------------------------------
__global__ void zero_f32(float* __restrict__ p, int n) {
  int i = blockIdx.x * blockDim.x + threadIdx.x;
  if (i < n) p[i] = 0.0f;
}

// generic scatter (used for conv1's 9-wide features)
__global__ void scatter_relu_add(const float* __restrict__ h,
                                 const int* __restrict__ src,
                                 const int* __restrict__ dst,
                                 float* __restrict__ agg,
                                 int E, int Df) {
  int tid = blockIdx.x * blockDim.x + threadIdx.x;
  int total = E * Df;
  if (tid >= total) return;
  int e = tid / Df;
  int f = tid - e * Df;
  if (f == 0 && e + 4096 < E) {
    __builtin_prefetch(&src[e + 4096], 0, 0);
    __builtin_prefetch(&dst[e + 4096], 0, 0);
  }
  float v = h[src[e] * Df + f];
  v = v > 0.0f ? v : 0.0f;
  atomicAdd(&agg[dst[e] * Df + f], v);
}

// vectorized scatter for 96-wide features: one float4 per thread
__global__ void scatter_relu_add_v4(const float* __restrict__ h,
                                    const int* __restrict__ src,
                                    const int* __restrict__ dst,
                                    float* __restrict__ agg, int E) {
  int tid = blockIdx.x * blockDim.x + threadIdx.x;
  int total = E * 24;               // 24 float4 per edge
  if (tid >= total) return;
  int e  = tid / 24;
  int c4 = tid - e * 24;
  if (c4 == 0 && e + 4096 < E) {
    __builtin_prefetch(&src[e + 4096], 0, 0);   // gfx1250 global_prefetch_b8
    __builtin_prefetch(&dst[e + 4096], 0, 0);
  }
  int sn = src[e], dn = dst[e];
  float4 v = ((const float4*)(h + (size_t)sn * DIMF))[c4];
  float* ap = agg + (size_t)dn * DIMF + c4 * 4;
  atomicAdd(ap + 0, fmaxf(v.x, 0.0f));
  atomicAdd(ap + 1, fmaxf(v.y, 0.0f));
  atomicAdd(ap + 2, fmaxf(v.z, 0.0f));
  atomicAdd(ap + 3, fmaxf(v.w, 0.0f));
}

// ------------------------------------------------------------------
// conv1: 9 -> 9 -> 96 (too small for WMMA, one node per thread)
// ------------------------------------------------------------------
__global__ void conv1_kernel(const float* __restrict__ x,
                             const float* __restrict__ agg,
                             const float* __restrict__ W1,   // 9x9
                             const float* __restrict__ b1,   // 9
                             const float* __restrict__ W2,   // 9x96
                             const float* __restrict__ b2,   // 96
                             const float* __restrict__ eps,
                             float* __restrict__ hout, int N) {
  int n = blockIdx.x * blockDim.x + threadIdx.x;
  if (n >= N) return;
  float epsv = 1.0f + eps[0];
  float hin[NFEAT], t[NFEAT];
#pragma unroll
  for (int k = 0; k < NFEAT; ++k)
    hin[k] = epsv * x[n * NFEAT + k] + agg[n * NFEAT + k];
#pragma unroll
  for (int j = 0; j < NFEAT; ++j) {
    float acc = b1[j];
#pragma unroll
    for (int k = 0; k < NFEAT; ++k) acc += hin[k] * W1[k * NFEAT + j];
    t[j] = acc > 0.0f ? acc : 0.0f;
  }
  for (int f = 0; f < DIMF; ++f) {
    float acc = b2[f];
#pragma unroll
    for (int j = 0; j < NFEAT; ++j) acc += t[j] * W2[j * DIMF + f];
    hout[n * DIMF + f] = acc > 0.0f ? acc : 0.0f;  // outer relu
  }
}

// ------------------------------------------------------------------
// GIN MLP for conv2..conv6: both 96x96 GEMMs fused per 16-node tile.
// Block = 192 threads = 6 wave32.  Weights are DMA'd into LDS by the
// Tensor Data Mover (overlapped with the activation tile load), WMMA
// B-operands then come from LDS instead of 96 global loads per wave.
// Dynamic LDS layout (floats): tin[16*100] | tmid[16*100] | W1[9216] | W2[9216]
// ------------------------------------------------------------------
#define TIN_OFF   0
#define TMID_OFF  1600
#define W1_OFF    3200
#define W2_OFF    12416
#define GIN_SMEM_FLOATS 21632
#define GIN_SMEM_BYTES  (GIN_SMEM_FLOATS * 4)

__global__ __launch_bounds__(192) void gin_mlp_wmma(
    const float* __restrict__ h, const float* __restrict__ agg,
    const float* __restrict__ eps,
    const float* __restrict__ W1, const float* __restrict__ b1,
    const float* __restrict__ W2, const float* __restrict__ b2,
    float* __restrict__ hout, int N)
{
  extern __shared__ float smem[];
  float* tin  = smem + TIN_OFF;     // [16][100]
  float* tmid = smem + TMID_OFF;    // [16][100]
  float* w1l  = smem + W1_OFF;      // [96][96]
  float* w2l  = smem + W2_OFF;      // [96][96]

  const int tile0 = blockIdx.x * 16;
  const float epsv = 1.0f + eps[0];
  const int wave  = threadIdx.x >> 5;

#if HAVE_TDM
  // wave 0 kicks off both weight DMAs before touching the tile
  if (wave == 0) {
    tdm_load_w96(W1, W1_OFF * 4);
    tdm_load_w96(W2, W2_OFF * 4);
  }
#endif

  // cooperative load of (1+eps)*h + agg, 16 rows x 24 float4 (overlaps TDM)
  for (int i = 0; i < 2; ++i) {
    int idx = threadIdx.x + i * 192;        // 0..383
    int row = idx / 24;
    int c4  = idx - row * 24;
    int node = tile0 + row;
    float vx = 0.f, vy = 0.f, vz = 0.f, vw = 0.f;
    if (node < N) {
      const float4 hv = ((const float4*)(h   + (size_t)node * DIMF))[c4];
      const float4 av = ((const float4*)(agg + (size_t)node * DIMF))[c4];
      vx = epsv * hv.x + av.x; vy = epsv * hv.y + av.y;
      vz = epsv * hv.z + av.z; vw = epsv * hv.w + av.w;
    }
    float* tr = tin + row * 100 + c4 * 4;
    tr[0] = vx; tr[1] = vy; tr[2] = vz; tr[3] = vw;
  }

#if HAVE_TDM
  if (wave == 0) __builtin_amdgcn_s_wait_tensorcnt(0);  // TDM writes visible
#else
  for (int i = threadIdx.x; i < 96 * 96; i += 192) { w1l[i] = W1[i]; w2l[i] = W2[i]; }
#endif
  __syncthreads();

  const int lane  = threadIdx.x & 31;
  const int lrow  = lane & 15;
  const int khalf = (lane >> 4) << 1;   // 0 or 2 (A/B K-pair select)
  const int chi   = lane >> 4;          // 0/1 (C/D row-half select)
  const int ncol  = wave * 16 + lrow;   // 0..95

  // GEMM1: tin(16x96) @ W1(96x96) -> tmid, +b1, relu
  v8f acc = {0, 0, 0, 0, 0, 0, 0, 0};
#pragma unroll 4
  for (int kb = 0; kb < 24; ++kb) {
    int k = kb * 4 + khalf;
    v2f a; a.x = tin[lrow * 100 + k];   a.y = tin[lrow * 100 + k + 1];
    v2f b; b.x = w1l[k * DIMF + ncol];  b.y = w1l[(k + 1) * DIMF + ncol];
    acc = wmma_f32_16x16x4(a, b, acc);
  }
  {
    float bias = b1[ncol];
#pragma unroll
    for (int r = 0; r < 8; ++r) {
      float v = acc[r] + bias;
      tmid[(r + 8 * chi) * 100 + ncol] = v > 0.0f ? v : 0.0f;
    }
  }
  __syncthreads();

  // GEMM2: tmid(16x96) @ W2(96x96), +b2, outer relu, store
  v8f acc2 = {0, 0, 0, 0, 0, 0, 0, 0};
#pragma unroll 4
  for (int kb = 0; kb < 24; ++kb) {
    int k = kb * 4 + khalf;
    v2f a; a.x = tmid[lrow * 100 + k];  a.y = tmid[lrow * 100 + k + 1];
    v2f b; b.x = w2l[k * DIMF + ncol];  b.y = w2l[(k + 1) * DIMF + ncol];
    acc2 = wmma_f32_16x16x4(a, b, acc2);
  }
  {
    float bias = b2[ncol];
#pragma unroll
    for (int r = 0; r < 8; ++r) {
      int node = tile0 + r + 8 * chi;
      if (node < N) {
        float v = acc2[r] + bias;
        hout[(size_t)node * DIMF + ncol] = v > 0.0f ? v : 0.0f;
      }
    }
  }
}

// ------------------------------------------------------------------
// Set2Set LSTM gate GEMM: [B,288] @ [288,384] (Wih^T | Whh^T) + bih + bhh
// grid = (ceil(B/16), 4), 6 waves cover 96 cols per block.
// ------------------------------------------------------------------
#define KCAT 288
#define QSTRIDE 292
__global__ __launch_bounds__(192) void gates_wmma(
    const float* __restrict__ qcat,   // [B,288] = [q_star | h]
    const float* __restrict__ Wih,    // [384,192]
    const float* __restrict__ Whh,    // [384,96]
    const float* __restrict__ bih, const float* __restrict__ bhh,
    float* __restrict__ gates, int B)
{
  __shared__ float tq[16][QSTRIDE];
  const int tile0 = blockIdx.x * 16;
  for (int i = 0; i < 6; ++i) {
    int idx = threadIdx.x + i * 192;  // 0..1151 (16 x 72 float4)
    int row = idx / 72;
    int c4  = idx - row * 72;
    int b = tile0 + row;
    float4 v = {0.f, 0.f, 0.f, 0.f};
    if (b < B) v = ((const float4*)(qcat + (size_t)b * KCAT))[c4];
    tq[row][c4 * 4 + 0] = v.x; tq[row][c4 * 4 + 1] = v.y;
    tq[row][c4 * 4 + 2] = v.z; tq[row][c4 * 4 + 3] = v.w;
  }
  __syncthreads();

  const int wave  = threadIdx.x >> 5;
  const int lane  = threadIdx.x & 31;
  const int lrow  = lane & 15;
  const int khalf = (lane >> 4) << 1;
  const int chi   = lane >> 4;
  const int ncol  = blockIdx.y * 96 + wave * 16 + lrow;   // 0..383

  v8f acc = {0, 0, 0, 0, 0, 0, 0, 0};
  for (int kb = 0; kb < 72; ++kb) {
    int k = kb * 4 + khalf;
    v2f a; a.x = tq[lrow][k]; a.y = tq[lrow][k + 1];
    v2f b;
    if (kb < 48) { // K in [0,192): q_star @ Wih^T  -> B[k][n] = Wih[n][k]
      b.x = Wih[(size_t)ncol * 192 + k];
      b.y = Wih[(size_t)ncol * 192 + k + 1];
    } else {       // K in [192,288): h @ Whh^T -> B[k][n] = Whh[n][k-192]
      b.x = Whh[(size_t)ncol * 96 + (k - 192)];
      b.y = Whh[(size_t)ncol * 96 + (k - 191)];
    }
    acc = wmma_f32_16x16x4(a, b, acc);
  }
  float bias = bih[ncol] + bhh[ncol];
#pragma unroll
  for (int r = 0; r < 8; ++r) {
    int b = tile0 + r + 8 * chi;
    if (b < B) gates[(size_t)b * 384 + ncol] = acc[r] + bias;
  }
}

// ------------------------------------------------------------------
// fc1: relu(q_star[B,192] @ W[192,96] + b)
// ------------------------------------------------------------------
#define FSTRIDE 196
__global__ __launch_bounds__(192) void fc1_wmma(
    const float* __restrict__ qstar,  // [B,192]
    const float* __restrict__ W,      // [192,96] row-major (already B[k][n])
    const float* __restrict__ bias,   // [96]
    float* __restrict__ z, int B)
{
  __shared__ float tq[16][FSTRIDE];
  const int tile0 = blockIdx.x * 16;
  for (int i = 0; i < 4; ++i) {
    int idx = threadIdx.x + i * 192;  // 0..767 (16 x 48 float4)
    int row = idx / 48;
    int c4  = idx - row * 48;
    int b = tile0 + row;
    float4 v = {0.f, 0.f, 0.f, 0.f};
    if (b < B) v = ((const float4*)(qstar + (size_t)b * 192))[c4];
    tq[row][c4 * 4 + 0] = v.x; tq[row][c4 * 4 + 1] = v.y;
    tq[row][c4 * 4 + 2] = v.z; tq[row][c4 * 4 + 3] = v.w;
  }
  __syncthreads();

  const int wave  = threadIdx.x >> 5;
  const int lane  = threadIdx.x & 31;
  const int lrow  = lane & 15;
  const int khalf = (lane >> 4) << 1;
  const int chi   = lane >> 4;
  const int ncol  = wave * 16 + lrow;   // 0..95

  v8f acc = {0, 0, 0, 0, 0, 0, 0, 0};
#pragma unroll 4
  for (int kb = 0; kb < 48; ++kb) {
    int k = kb * 4 + khalf;
    v2f a; a.x = tq[lrow][k];        a.y = tq[lrow][k + 1];
    v2f b; b.x = W[k * 96 + ncol];   b.y = W[(k + 1) * 96 + ncol];
    acc = wmma_f32_16x16x4(a, b, acc);
  }
  float bv = bias[ncol];
#pragma unroll
  for (int r = 0; r < 8; ++r) {
    int b = tile0 + r + 8 * chi;
    if (b < B) {
      float v = acc[r] + bv;
      z[(size_t)b * 96 + ncol] = v > 0.0f ? v : 0.0f;
    }
  }
}

// ------------------------------------------------------------------
// pointwise / segment kernels for Set2Set
// ------------------------------------------------------------------
__global__ void lstm_cell(const float* __restrict__ gates,
                          float* __restrict__ c, float* __restrict__ h, int B) {
  int tid = blockIdx.x * blockDim.x + threadIdx.x;
  if (tid >= B * DIMF) return;
  int b = tid / DIMF, f = tid - b * DIMF;
  const float* g = gates + (size_t)b * 384;
  float ig = 1.0f / (1.0f + expf(-g[f]));
  float fg = 1.0f / (1.0f + expf(-g[96 + f]));
  float gg = tanhf(g[192 + f]);
  float og = 1.0f / (1.0f + expf(-g[288 + f]));
  float cv = fg * c[tid] + ig * gg;
  c[tid] = cv;
  h[tid] = og * tanhf(cv);
}

__global__ void build_qcat(const float* __restrict__ qstar,
                           const float* __restrict__ hs,
                           float* __restrict__ qcat, int B) {
  int tid = blockIdx.x * blockDim.x + threadIdx.x;
  if (tid >= B * 3 * DIMF) return;
  int b = tid / (3 * DIMF);
  int c = tid - b * 3 * DIMF;
  qcat[tid] = (c < 2 * DIMF) ? qstar[(size_t)b * 2 * DIMF + c]
                             : hs[(size_t)b * DIMF + (c - 2 * DIMF)];
}

__global__ void attn_e(const float* __restrict__ hn, const float* __restrict__ hs,
                       const int* __restrict__ batch, float* __restrict__ e, int N) {
  int n = blockIdx.x * blockDim.x + threadIdx.x;
  if (n >= N) return;
  int b = batch[n];
  const float* xp = hn + (size_t)n * DIMF;
  const float* qp = hs + (size_t)b * DIMF;
  float acc = 0.0f;
#pragma unroll 8
  for (int f = 0; f < DIMF; ++f) acc += xp[f] * qp[f];
  e[n] = acc;
}

__global__ void init_ms(float* __restrict__ m, float* __restrict__ s, int B) {
  int i = blockIdx.x * blockDim.x + threadIdx.x;
  if (i < B) { m[i] = -3.402823466e38f; s[i] = 0.0f; }
}

__device__ __forceinline__ void atomicMaxF(float* addr, float val) {
  unsigned int* ua = (unsigned int*)addr;
  unsigned int old = __float_as_uint(*addr);
  while (__uint_as_float(old) < val) {
    unsigned int assumed = old;
    old = atomicCAS(ua, assumed, __float_as_uint(val));
    if (old == assumed) break;
  }
}

__global__ void seg_max(const float* __restrict__ e, const int* __restrict__ batch,
                        float* __restrict__ m, int N) {
  int n = blockIdx.x * blockDim.x + threadIdx.x;
  if (n >= N) return;
  atomicMaxF(&m[batch[n]], e[n]);
}

__global__ void seg_expsum(float* __restrict__ e, const float* __restrict__ m,
                           const int* __restrict__ batch, float* __restrict__ s, int N) {
  int n = blockIdx.x * blockDim.x + threadIdx.x;
  if (n >= N) return;
  int b = batch[n];
  float ex = expf(e[n] - m[b]);
  e[n] = ex;
  atomicAdd(&s[b], ex);
}

__global__ void seg_r(const float* __restrict__ ex, const float* __restrict__ s,
                      const int* __restrict__ batch, const float* __restrict__ hn,
                      float* __restrict__ r, int N) {
  int tid = blockIdx.x * blockDim.x + threadIdx.x;
  if (tid >= N * DIMF) return;
  int n = tid / DIMF, f = tid - n * DIMF;
  int b = batch[n];
  float a = ex[n] / s[b];
  atomicAdd(&r[(size_t)b * DIMF + f], a * hn[tid]);
}

__global__ void qstar_upd(const float* __restrict__ hs, const float* __restrict__ r,
                          float* __restrict__ qstar, int B) {
  int tid = blockIdx.x * blockDim.x + threadIdx.x;
  if (tid >= B * DIMF) return;
  int b = tid / DIMF, f = tid - b * DIMF;
  qstar[(size_t)b * 2 * DIMF + f] = hs[tid];
  qstar[(size_t)b * 2 * DIMF + DIMF + f] = r[tid];
}

__global__ void fc4_kernel(const float* __restrict__ z, const float* __restrict__ W,
                           const float* __restrict__ bias, float* __restrict__ out, int B) {
  int tid = blockIdx.x * blockDim.x + threadIdx.x;
  if (tid >= B * 12) return;
  int b = tid / 12, j = tid - b * 12;
  float acc = bias[j];
#pragma unroll 8
  for (int f = 0; f < DIMF; ++f) acc += z[(size_t)b * DIMF + f] * W[f * 12 + j];
  out[tid] = acc;
}

// ------------------------------------------------------------------
// launch sequence
// ------------------------------------------------------------------
extern "C" void kernel_launch(void* const* d_in, const int* in_sizes, int n_in,
                              void* d_out, int out_size, void* d_ws, size_t ws_size,
                              hipStream_t stream) {
  (void)in_sizes; (void)n_in; (void)out_size; (void)ws_size;
  const int N = 50000, E = 800000, B = NGRAPH;

  const float* x     = (const float*)d_in[0];
  const int*   ei    = (const int*)d_in[1];
  const int*   src   = ei;
  const int*   dst   = ei + E;
  const int*   batch = (const int*)d_in[2];

  // params: pytree sorted-key order:
  // conv1..conv6 {W1,W2,b1,b2,eps}, fc1 {W,b}, fc4 {W,b}, lstm {Whh,Wih,bhh,bih}
  int p = 3;
  const float *cW1[6], *cW2[6], *cb1[6], *cb2[6], *ceps[6];
  for (int i = 0; i < 6; ++i) {
    cW1[i]  = (const float*)d_in[p++];
    cW2[i]  = (const float*)d_in[p++];
    cb1[i]  = (const float*)d_in[p++];
    cb2[i]  = (const float*)d_in[p++];
    ceps[i] = (const float*)d_in[p++];
  }
  const float* fc1W = (const float*)d_in[p++];
  const float* fc1b = (const float*)d_in[p++];
  const float* fc4W = (const float*)d_in[p++];
  const float* fc4b = (const float*)d_in[p++];
  const float* Whh  = (const float*)d_in[p++];
  const float* Wih  = (const float*)d_in[p++];
  const float* bhh  = (const float*)d_in[p++];
  const float* bih  = (const float*)d_in[p++];

  // workspace carve (~58 MB of fp32)
  float* w = (float*)d_ws;
  float* hA    = w; w += (size_t)N * DIMF;
  float* hB    = w; w += (size_t)N * DIMF;
  float* agg   = w; w += (size_t)N * DIMF;
  float* e     = w; w += N;
  float* m     = w; w += B;
  float* s     = w; w += B;
  float* r     = w; w += (size_t)B * DIMF;
  float* hs    = w; w += (size_t)B * DIMF;
  float* cs    = w; w += (size_t)B * DIMF;
  float* qstar = w; w += (size_t)B * 2 * DIMF;
  float* qcat  = w; w += (size_t)B * 3 * DIMF;
  float* gates = w; w += (size_t)B * 4 * DIMF;
  float* z     = w; w += (size_t)B * DIMF;

  // ---- conv1 (9 -> 96) ----
  zero_f32<<<(N * NFEAT + 255) / 256, 256, 0, stream>>>(agg, N * NFEAT);
  scatter_relu_add<<<(E * NFEAT + 255) / 256, 256, 0, stream>>>(x, src, dst, agg, E, NFEAT);
  conv1_kernel<<<(N + 127) / 128, 128, 0, stream>>>(x, agg, cW1[0], cb1[0], cW2[0], cb2[0],
                                                    ceps[0], hA, N);

  // ---- conv2..conv6 (96 -> 96), TDM-staged weights + WMMA MLP ----
  float* cur = hA; float* nxt = hB;
  const int mtiles = (N + 15) / 16;   // 3125
  for (int L = 1; L < 6; ++L) {
    zero_f32<<<(N * DIMF + 255) / 256, 256, 0, stream>>>(agg, N * DIMF);
    scatter_relu_add_v4<<<(E * 24 + 255) / 256, 256, 0, stream>>>(cur, src, dst, agg, E);
    gin_mlp_wmma<<<mtiles, 192, GIN_SMEM_BYTES, stream>>>(cur, agg, ceps[L], cW1[L], cb1[L],
                                                          cW2[L], cb2[L], nxt, N);
    float* t = cur; cur = nxt; nxt = t;
  }
  const float* hn = cur;   // final node embeddings [N,96]

  // ---- Set2Set ----
  zero_f32<<<(B * DIMF + 255) / 256, 256, 0, stream>>>(hs, B * DIMF);
  zero_f32<<<(B * DIMF + 255) / 256, 256, 0, stream>>>(cs, B * DIMF);
  zero_f32<<<(B * 2 * DIMF + 255) / 256, 256, 0, stream>>>(qstar, B * 2 * DIMF);

  const int btiles = (B + 15) / 16;  // 63
  for (int step = 0; step < S2S_STEPS; ++step) {
    build_qcat<<<(B * 3 * DIMF + 255) / 256, 256, 0, stream>>>(qstar, hs, qcat, B);
    gates_wmma<<<dim3(btiles, 4), 192, 0, stream>>>(qcat, Wih, Whh, bih, bhh, gates, B);
    lstm_cell<<<(B * DIMF + 255) / 256, 256, 0, stream>>>(gates, cs, hs, B);
    attn_e<<<(N + 127) / 128, 128, 0, stream>>>(hn, hs, batch, e, N);
    init_ms<<<(B + 255) / 256, 256, 0, stream>>>(m, s, B);
    zero_f32<<<(B * DIMF + 255) / 256, 256, 0, stream>>>(r, B * DIMF);
    seg_max<<<(N + 255) / 256, 256, 0, stream>>>(e, batch, m, N);
    seg_expsum<<<(N + 255) / 256, 256, 0, stream>>>(e, m, batch, s, N);
    seg_r<<<(N * DIMF + 255) / 256, 256, 0, stream>>>(e, s, batch, hn, r, N);
    qstar_upd<<<(B * DIMF + 255) / 256, 256, 0, stream>>>(hs, r, qstar, B);
  }

  // ---- output head ----
  fc1_wmma<<<btiles, 192, 0, stream>>>(qstar, fc1W, fc1b, z, B);
  fc4_kernel<<<(B * 12 + 255) / 256, 256, 0, stream>>>(z, fc4W, fc4b, (float*)d_out, B);
}